// BaseLine_53334903882347
// MI455X (gfx1250) — compile-verified
//
#include <hip/hip_runtime.h>

#define N_NODES 100000
#define N_EDGES 1600000
#define IN_DIM  512
#define HID     32
#define BN_EPS  1e-5f

typedef __attribute__((ext_vector_type(2))) float v2f;
typedef __attribute__((ext_vector_type(8))) float v8f;

// ---------------- utility fills ----------------
__global__ void gcn_fill_kernel(float* __restrict__ p, float v, int n) {
    int i = blockIdx.x * blockDim.x + threadIdx.x;
    if (i < n) p[i] = v;
}

// deg[dst] += 1  (deg pre-filled with 1.0 for the self loop)
__global__ void gcn_deg_kernel(const int* __restrict__ dst, float* __restrict__ deg) {
    int e = blockIdx.x * blockDim.x + threadIdx.x;
    if (e < N_EDGES)
        __hip_atomic_fetch_add(&deg[dst[e]], 1.0f, __ATOMIC_RELAXED, __HIP_MEMORY_SCOPE_AGENT);
}

// deg -> dinv in place
__global__ void gcn_dinv_kernel(float* __restrict__ deg) {
    int i = blockIdx.x * blockDim.x + threadIdx.x;
    if (i < N_NODES) deg[i] = rsqrtf(deg[i]);
}

// ---------------- dense GEMM via V_WMMA_F32_16X16X4_F32 ----------------
// h[N,32] = x[N,K] @ W[K,32]. One wave owns TWO 16-row tiles (a 32x32 output
// block, 4 accumulators) so the B fragments (shared W rows) are loaded once
// per K-step for 4 WMMAs instead of 2 -> fewer vmem issues per HBM byte.
// A layout (16x4 f32): lane = {M = lane&15, Kpair = (lane>>4)*2}
// B layout (4x16 f32): lane = {N = lane&15, Kpair = (lane>>4)*2}
// C/D layout: VGPR v, lane L -> D[v + 8*(L>>4)][L&15]
template <int K>
__global__ __launch_bounds__(256) void gcn_gemm_kernel(const float* __restrict__ x,
                                                       const float* __restrict__ W,
                                                       float* __restrict__ h) {
    const int wave = threadIdx.x >> 5;
    const int lane = threadIdx.x & 31;
    const int pair = blockIdx.x * 8 + wave;        // 32-row block index (wave-uniform)
    if (pair >= N_NODES / 32) return;              // 100000/32 = 3125 exact; EXEC all-1s
    const int row0 = pair * 32;
    const int m  = lane & 15;
    const int kk = (lane >> 4) << 1;               // 0 or 2: K-pair this half-wave owns

    const float* xr0 = x + (size_t)(row0 + m) * K + kk;        // tile 0: rows row0..row0+15
    const float* xr1 = xr0 + (size_t)16 * K;                    // tile 1: rows row0+16..+31

    v8f c00 = {};  // tile0, cols 0..15
    v8f c01 = {};  // tile0, cols 16..31
    v8f c10 = {};  // tile1, cols 0..15
    v8f c11 = {};  // tile1, cols 16..31
#pragma unroll 4
    for (int k = 0; k < K; k += 4) {
        v2f a0 = *(const v2f*)(xr0 + k);
        v2f a1 = *(const v2f*)(xr1 + k);
        const float* wr = W + (size_t)(k + kk) * HID;   // row k+kk of W (32 floats)
        v2f b0; b0.x = wr[m];       b0.y = wr[HID + m];        // cols 0..15
        v2f b1; b1.x = wr[16 + m];  b1.y = wr[HID + 16 + m];   // cols 16..31
        c00 = __builtin_amdgcn_wmma_f32_16x16x4_f32(false, a0, false, b0, (short)0, c00, false, false);
        c01 = __builtin_amdgcn_wmma_f32_16x16x4_f32(false, a0, false, b1, (short)0, c01, false, false);
        c10 = __builtin_amdgcn_wmma_f32_16x16x4_f32(false, a1, false, b0, (short)0, c10, false, false);
        c11 = __builtin_amdgcn_wmma_f32_16x16x4_f32(false, a1, false, b1, (short)0, c11, false, false);
    }

    const int rb0 = row0 + ((lane >> 4) << 3);     // tile0 row base: +0 or +8
    const int rb1 = rb0 + 16;                      // tile1 row base
#pragma unroll
    for (int v = 0; v < 8; ++v) {
        h[(size_t)(rb0 + v) * HID + m]      = c00[v];
        h[(size_t)(rb0 + v) * HID + 16 + m] = c01[v];
        h[(size_t)(rb1 + v) * HID + m]      = c10[v];
        h[(size_t)(rb1 + v) * HID + 16 + m] = c11[v];
    }
}

// ---------------- edge scatter: agg[dst] += h[src] * dinv[src]*dinv[dst] ----------------
// One wave per edge, lane = feature (wave32 == HID, perfectly coalesced).
// Edge index / dinv fetches are forced wave-uniform -> SMEM scalar loads.
__global__ void gcn_scatter_kernel(const float* __restrict__ h,
                                   const int* __restrict__ src,
                                   const int* __restrict__ dst,
                                   const float* __restrict__ dinv,
                                   float* __restrict__ agg) {
    const int wave = __builtin_amdgcn_readfirstlane(threadIdx.x >> 5);  // uniform
    const int lane = threadIdx.x & 31;
    const int e = blockIdx.x * 8 + wave;           // 8 waves per 256-thread block
    if (e >= N_EDGES) return;
    const int s = src[e];                          // uniform -> s_load
    const int d = dst[e];
    const float coef = dinv[s] * dinv[d];
    float v = h[(size_t)s * HID + lane] * coef;    // coalesced 128B per wave (L2 hit)
    __hip_atomic_fetch_add(&agg[(size_t)d * HID + lane], v,
                           __ATOMIC_RELAXED, __HIP_MEMORY_SCOPE_AGENT);
}

// ---------------- epilogue 1: self-loop + bias + ReLU + BatchNorm(eval) ----------------
__global__ void gcn_epilogue1_kernel(const float* __restrict__ agg,
                                     const float* __restrict__ h,
                                     const float* __restrict__ dinv,
                                     const float* __restrict__ b,
                                     const float* __restrict__ gamma,
                                     const float* __restrict__ beta,
                                     const float* __restrict__ mean,
                                     const float* __restrict__ var,
                                     float* __restrict__ out) {
    int t = blockIdx.x * blockDim.x + threadIdx.x;
    if (t >= N_NODES * HID) return;
    int n = t >> 5;
    int j = t & 31;
    float di = dinv[n];
    float v = agg[t] + h[t] * di * di + b[j];
    v = fmaxf(v, 0.0f);
    v = (v - mean[j]) * rsqrtf(var[j] + BN_EPS) * gamma[j] + beta[j];
    out[t] = v;
}

// ---------------- epilogue 2: self-loop + bias + ReLU ----------------
__global__ void gcn_epilogue2_kernel(const float* __restrict__ agg,
                                     const float* __restrict__ h,
                                     const float* __restrict__ dinv,
                                     const float* __restrict__ b,
                                     float* __restrict__ out) {
    int t = blockIdx.x * blockDim.x + threadIdx.x;
    if (t >= N_NODES * HID) return;
    int n = t >> 5;
    int j = t & 31;
    float di = dinv[n];
    float v = agg[t] + h[t] * di * di + b[j];
    out[t] = fmaxf(v, 0.0f);
}

extern "C" void kernel_launch(void* const* d_in, const int* in_sizes, int n_in,
                              void* d_out, int out_size, void* d_ws, size_t ws_size,
                              hipStream_t stream) {
    const float* x     = (const float*)d_in[0];
    const int*   ei    = (const int*)d_in[1];      // [2, N_EDGES]
    const int*   src   = ei;
    const int*   dst   = ei + N_EDGES;
    const float* W1    = (const float*)d_in[2];
    const float* b1    = (const float*)d_in[3];
    const float* W2    = (const float*)d_in[4];
    const float* b2    = (const float*)d_in[5];
    const float* gamma = (const float*)d_in[6];
    const float* beta  = (const float*)d_in[7];
    const float* mean  = (const float*)d_in[8];
    const float* var   = (const float*)d_in[9];

    const size_t NF = (size_t)N_NODES * HID;       // 3.2M floats per node-feature buffer
    float* deg  = (float*)d_ws;                    // N       (becomes dinv)
    float* h1   = deg  + N_NODES;                  // N*32
    float* agg1 = h1   + NF;                       // N*32    (becomes layer-2 input in place)
    float* h2   = agg1 + NF;                       // N*32
    float* agg2 = h2   + NF;                       // N*32    total ~51.6 MB

    const int T = 256;
    const int nNodeBlk = (N_NODES + T - 1) / T;
    const int nFeatBlk = (int)((NF + T - 1) / T);
    const int nEdgeBlk = (N_EDGES + T - 1) / T;
    const int nScatBlk = (N_EDGES + 7) / 8;        // 8 edges (waves) per block
    const int nGemmBlk = (N_NODES / 32 + 7) / 8;   // 8 waves/block, 32 rows/wave

    // init: deg = 1 (self loop), agg buffers = 0 (must re-init every call)
    gcn_fill_kernel<<<nNodeBlk, T, 0, stream>>>(deg, 1.0f, N_NODES);
    gcn_fill_kernel<<<nFeatBlk, T, 0, stream>>>(agg1, 0.0f, (int)NF);
    gcn_fill_kernel<<<nFeatBlk, T, 0, stream>>>(agg2, 0.0f, (int)NF);

    // degrees -> dinv
    gcn_deg_kernel<<<nEdgeBlk, T, 0, stream>>>(dst, deg);
    gcn_dinv_kernel<<<nNodeBlk, T, 0, stream>>>(deg);

    // layer 1
    gcn_gemm_kernel<IN_DIM><<<nGemmBlk, T, 0, stream>>>(x, W1, h1);
    gcn_scatter_kernel<<<nScatBlk, T, 0, stream>>>(h1, src, dst, deg, agg1);
    gcn_epilogue1_kernel<<<nFeatBlk, T, 0, stream>>>(agg1, h1, deg, b1, gamma, beta,
                                                     mean, var, agg1);  // in-place -> x2

    // layer 2
    gcn_gemm_kernel<HID><<<nGemmBlk, T, 0, stream>>>(agg1, W2, h2);
    gcn_scatter_kernel<<<nScatBlk, T, 0, stream>>>(h2, src, dst, deg, agg2);
    gcn_epilogue2_kernel<<<nFeatBlk, T, 0, stream>>>(agg2, h2, deg, b2, (float*)d_out);
}